// MultiChannelXi_18202071400698
// MI455X (gfx1250) — compile-verified
//
#include <hip/hip_runtime.h>
#include <cmath>

typedef float v2f __attribute__((ext_vector_type(2)));
typedef float v8f __attribute__((ext_vector_type(8)));

namespace {

constexpr int   B_SZ   = 4;
constexpr int   T_LEN  = 4096;
constexpr int   D_DIM  = 512;
constexpr int   K_EMA  = 4;
constexpr int   CHUNK  = 16;              // == WMMA M; chunked-scan block
constexpr int   G_CH   = T_LEN / CHUNK;   // 256 chunks
constexpr float ALPHA_EPS = 1e-6f;

__device__ __forceinline__ float sigmoid_clip(float x) {
  float a = 1.0f / (1.0f + expf(-x));
  return fminf(fmaxf(a, ALPHA_EPS), 1.0f - ALPHA_EPS);
}

// ---------------------------------------------------------------------------
// Pass 1: per-chunk local tail sums  L[b,k,g,c] = sum_{s=0..15} a_k^{15-s} h[b,gC+s,c]
// Fully parallel over (b,g,c); reads h exactly once for all K channels.
// ---------------------------------------------------------------------------
__global__ void ema_chunk_tails(const float* __restrict__ h,
                                const float* __restrict__ raw_alpha,
                                float* __restrict__ Lt) {
  int idx = blockIdx.x * blockDim.x + threadIdx.x;
  if (idx >= B_SZ * G_CH * D_DIM) return;
  int c = idx % D_DIM;
  int g = (idx / D_DIM) % G_CH;
  int b = idx / (D_DIM * G_CH);

  float al[K_EMA];
#pragma unroll
  for (int k = 0; k < K_EMA; ++k) al[k] = sigmoid_clip(raw_alpha[k]);

  float tail[K_EMA] = {0.f, 0.f, 0.f, 0.f};
  const float* hp = h + ((long)b * T_LEN + (long)g * CHUNK) * D_DIM + c;
#pragma unroll
  for (int s = 0; s < CHUNK; ++s) {
    float hv = hp[(long)s * D_DIM];
#pragma unroll
    for (int k = 0; k < K_EMA; ++k) tail[k] = al[k] * tail[k] + hv;
  }
#pragma unroll
  for (int k = 0; k < K_EMA; ++k)
    Lt[(((long)b * K_EMA + k) * G_CH + g) * D_DIM + c] = tail[k];
}

// ---------------------------------------------------------------------------
// Pass 2: scan over chunks.  S[b,k,g,c] = unnormalized EMA state entering
// chunk g (value at t = g*C-1); E_g = a^16 * E_{g-1} + L_{g-1}.
// 8192 independent scans of length 256 over 16 MiB of workspace.
// ---------------------------------------------------------------------------
__global__ void ema_chunk_scan(const float* __restrict__ Lt,
                               const float* __restrict__ raw_alpha,
                               float* __restrict__ St) {
  int idx = blockIdx.x * blockDim.x + threadIdx.x;
  if (idx >= B_SZ * K_EMA * D_DIM) return;
  int c = idx % D_DIM;
  int k = (idx / D_DIM) % K_EMA;
  int b = idx / (D_DIM * K_EMA);

  float alpha = sigmoid_clip(raw_alpha[k]);
  float a2 = alpha * alpha, a4 = a2 * a2, a8 = a4 * a4, a16 = a8 * a8;

  long base = ((long)b * K_EMA + k) * G_CH * D_DIM + c;
  float E = 0.f;
  for (int g = 0; g < G_CH; ++g) {
    St[base + (long)g * D_DIM] = E;
    E = a16 * E + Lt[base + (long)g * D_DIM];
  }
}

// ---------------------------------------------------------------------------
// Pass 3: one wave per 16x16 (time-rows x channels) output tile.
//   D = W_chunk(16x16 lower-tri alpha powers) @ H_chunk(16x16)   via
//   4 chained V_WMMA_F32_16X16X4_F32, then add decayed carry state and
//   normalize by Z_t = (1 - a^{t+1}) / (1 - a), with a^{t+1} factored as
//   a^{gC} * a^{m+1} so only ONE expf is needed per tile (not per row).
// A 16x4 f32 layout: a[v] = W[m=lane&15][k = v + 2*(lane>>4)]
// B  4x16 f32 layout: b[v] = H[k = v + 2*(lane>>4)][n = lane&15]
// C/D 16x16 f32:      d[r] = D[m = r + 8*(lane>>4)][n = lane&15]
// ---------------------------------------------------------------------------
__global__ void ema_wmma_tiles(const float* __restrict__ h,
                               const float* __restrict__ raw_alpha,
                               const float* __restrict__ St,
                               float* __restrict__ out) {
  const int lane = threadIdx.x & 31;
  const int wv   = threadIdx.x >> 5;
  const long tile = (long)blockIdx.x * (blockDim.x >> 5) + wv;

  const int NCT = D_DIM / 16;  // 32 channel tiles
  int ct = (int)(tile % NCT);
  int g  = (int)((tile / NCT) % G_CH);
  int k  = (int)((tile / ((long)NCT * G_CH)) % K_EMA);
  int b  = (int)(tile / ((long)NCT * G_CH * K_EMA));

  float alpha = sigmoid_clip(raw_alpha[k]);
  float l1a   = logf(alpha);
  // lane i holds alpha^i; alpha^j is fetched cross-lane (wave32 shuffle).
  float p  = expf(l1a * (float)lane);
  // alpha^(g*CHUNK): the only per-tile transcendental for the normalizer.
  float ag = expf(l1a * (float)(g * CHUNK));
  float one_m_a = 1.0f - alpha;

  const int hi = lane >> 4;
  const int mn = lane & 15;  // A row / B,C,D column index

  const float* hb =
      h + ((long)b * T_LEN + (long)g * CHUNK) * D_DIM + ct * 16 + mn;

  v8f acc = {};
#pragma unroll
  for (int j = 0; j < 4; ++j) {
    v2f a, bf;
#pragma unroll
    for (int v = 0; v < 2; ++v) {
      int s    = 4 * j + 2 * hi + v;  // K index within chunk, 0..15
      int diff = mn - s;              // causal: contribute only if s <= m
      float w  = __shfl(p, diff & 31);   // alpha^(m-s), all lanes participate
      a[v]  = (diff >= 0) ? w : 0.0f;
      bf[v] = hb[(long)s * D_DIM];
    }
    acc = __builtin_amdgcn_wmma_f32_16x16x4_f32(
        /*neg_a=*/false, a, /*neg_b=*/false, bf,
        /*c_mod=*/(short)0, acc, /*reuse_a=*/false, /*reuse_b=*/false);
  }

  // carry-in state for this chunk (one per channel)
  float S = St[(((long)b * K_EMA + k) * G_CH + g) * D_DIM + ct * 16 + mn];

  float* ob = out +
      (((long)b * T_LEN + (long)g * CHUNK) * K_EMA + k) * D_DIM + ct * 16 + mn;
#pragma unroll
  for (int r = 0; r < 8; ++r) {
    int   m     = r + 8 * hi;                 // time row within chunk
    float decay = __shfl(p, (m + 1) & 31);    // alpha^(m+1), m+1 <= 16
    float un    = acc[r] + decay * S;         // unnormalized EMA at t
    // Z_t = (1 - a^{t+1})/(1-a),  a^{t+1} = ag * decay
    float zden  = 1.0f - ag * decay;
    ob[(long)m * (K_EMA * D_DIM)] = un * one_m_a / zden;
  }
}

}  // namespace

extern "C" void kernel_launch(void* const* d_in, const int* in_sizes, int n_in,
                              void* d_out, int out_size, void* d_ws, size_t ws_size,
                              hipStream_t stream) {
  const float* h  = (const float*)d_in[0];   // (B, T, d) fp32
  const float* ra = (const float*)d_in[1];   // (K,) fp32
  float* out = (float*)d_out;                // (B, T, K, d) fp32

  // Workspace: Lt (chunk tails) and St (chunk-entry states), 8 MiB each.
  float* Lt = (float*)d_ws;
  float* St = Lt + (long)B_SZ * K_EMA * G_CH * D_DIM;

  {
    int n = B_SZ * G_CH * D_DIM;  // 524288
    ema_chunk_tails<<<(n + 255) / 256, 256, 0, stream>>>(h, ra, Lt);
  }
  {
    int n = B_SZ * K_EMA * D_DIM;  // 8192
    ema_chunk_scan<<<(n + 255) / 256, 256, 0, stream>>>(Lt, ra, St);
  }
  {
    long waves = (long)B_SZ * K_EMA * G_CH * (D_DIM / 16);  // 131072 tiles
    int  wpb   = 8;                                         // 8 waves = 256 thr
    ema_wmma_tiles<<<(int)(waves / wpb), wpb * 32, 0, stream>>>(h, ra, St, out);
  }
}